// SparseLinear_10866267259295
// MI455X (gfx1250) — compile-verified
//
#include <hip/hip_runtime.h>

#define IN_DIMS  8192
#define OUT_DIMS 8192

typedef __attribute__((ext_vector_type(4))) unsigned int v4u;
typedef __attribute__((ext_vector_type(8))) int          v8i;
typedef __attribute__((ext_vector_type(4))) int          v4i;
typedef __attribute__((ext_vector_type(4))) float        v4f;

// y = B  (fresh every launch; harness poisons d_out)
__global__ void SparseLinear_init_y(const float* __restrict__ B, float* __restrict__ y) {
    int i = blockIdx.x * blockDim.x + threadIdx.x;
    if (i < OUT_DIMS) y[i] = B[i];
}

__global__ __launch_bounds__(256) void SparseLinear_spmv_coo(
    const float* __restrict__ vals,
    const int*   __restrict__ rows,
    const int*   __restrict__ cols,
    const float* __restrict__ x,
    float*       __restrict__ y,
    int nnz)
{
    // x lives in LDS for the whole kernel: 8192 floats = 32 KB (<< 320 KB/WGP).
    __shared__ float xs[IN_DIMS];

    // ---- Stage x into LDS with one TDM tensor load, issued by wave 0 ----
    // D# group 0: count=1 (valid), lds_addr=0 (xs is the only LDS object),
    //             global_addr = &x, type=2 ("image").
    // D# group 1: data_size=2 (4 bytes), tensor_dim0=8192, tensor_dim1=1,
    //             tile_dim0=8192, tile_dim1=1, tensor_dim0_stride=8192.
    if (threadIdx.x < 32) {   // wave 0 only (wave32); TDM ignores EXEC
        unsigned long long xa = (unsigned long long)(const void*)x;
        v4u g0;
        g0.x = 1u;                                            // count=1, user descriptor
        g0.y = 0u;                                            // lds_addr = 0
        g0.z = (unsigned int)(xa & 0xFFFFFFFFu);              // global_addr[31:0]
        g0.w = (unsigned int)((xa >> 32) & 0x1FFFFFFu)        // global_addr[56:32]
             | (2u << 30);                                    // type = 2
        v8i g1;
        g1[0] = (int)(2u << 16);                              // data_size=4B, wg_mask=0
        g1[1] = (int)(IN_DIMS << 16);                         // tensor_dim0[15:0] @ [63:48]
        g1[2] = (int)(1u << 16);                              // tensor_dim0 hi=0, tensor_dim1=1
        g1[3] = (int)(IN_DIMS << 16);                         // tile_dim0 @ [127:112]
        g1[4] = 1;                                            // tile_dim1=1, tile_dim2=0
        g1[5] = IN_DIMS;                                      // tensor_dim0_stride[31:0]
        g1[6] = 0;
        g1[7] = 0;
        v4i gz4 = (v4i)0;                                     // groups 2/3 unused (<=2D)
        v8i gz8 = (v8i)0;                                     // 5th group (clang-23 6-arg form)
        __builtin_amdgcn_tensor_load_to_lds(g0, g1, gz4, gz4, gz8, 0);
        __builtin_amdgcn_s_wait_tensorcnt(0);                 // s_wait_tensorcnt 0
    }
    __syncthreads();                                          // publish LDS to all waves

    const int lane      = threadIdx.x & 31;
    const int waveInBlk = threadIdx.x >> 5;
    const long long wid    = (long long)blockIdx.x * (blockDim.x >> 5) + waveInBlk;
    const long long nWaves = (long long)gridDim.x * (blockDim.x >> 5);
    const long long nWin   = ((long long)nnz + 127) >> 7;     // 128 entries / wave / window

    for (long long w = wid; w < nWin; w += nWaves) {
        long long e0 = (w << 7) + (long long)lane * 4;        // 4 consecutive entries/lane

        int   r[4];
        float v[4];
        if (e0 + 3 < (long long)nnz) {
            // Coalesced 16B streaming loads, non-temporal (read-once data).
            v4i rr = __builtin_nontemporal_load((const v4i*)(rows + e0));
            v4i cc = __builtin_nontemporal_load((const v4i*)(cols + e0));
            v4f vv = __builtin_nontemporal_load((const v4f*)(vals + e0));
            r[0] = rr.x; r[1] = rr.y; r[2] = rr.z; r[3] = rr.w;
            v[0] = vv.x * xs[cc.x];                           // ds_load_b32 gathers
            v[1] = vv.y * xs[cc.y];
            v[2] = vv.z * xs[cc.z];
            v[3] = vv.w * xs[cc.w];
        } else {
            #pragma unroll
            for (int k = 0; k < 4; ++k) {
                long long idx = e0 + k;
                if (idx < (long long)nnz) {
                    r[k] = rows[idx];
                    v[k] = vals[idx] * xs[cols[idx]];
                } else {
                    r[k] = -1;                                // sentinel: never merges/flushes
                    v[k] = 0.0f;
                }
            }
        }

        // Intra-lane combine of the 4 contiguous entries (rows sorted -> runs).
        int   cur = r[0];
        float acc = v[0];
        #pragma unroll
        for (int k = 1; k < 4; ++k) {
            if (r[k] == cur) {
                acc += v[k];
            } else {
                if (cur >= 0) atomicAdd(&y[cur], acc);        // rare intra-quad break
                cur = r[k];
                acc = v[k];
            }
        }

        // Cross-lane segmented inclusive scan over the wave (wave32, width 32).
        int   key = cur;
        float val = acc;
        #pragma unroll
        for (int off = 1; off < 32; off <<= 1) {
            float up = __shfl_up(val, off, 32);
            int   ku = __shfl_up(key, off, 32);
            if (lane >= off && ku == key) val += up;
        }
        int  kd   = __shfl_down(key, 1, 32);
        bool tail = (lane == 31) || (kd != key);
        if (tail && key >= 0) atomicAdd(&y[key], val);        // one atomic per row-run
    }
}

extern "C" void kernel_launch(void* const* d_in, const int* in_sizes, int n_in,
                              void* d_out, int out_size, void* d_ws, size_t ws_size,
                              hipStream_t stream) {
    const float* x      = (const float*)d_in[0];
    const float* W_vals = (const float*)d_in[1];
    const int*   W_rows = (const int*)  d_in[2];
    const int*   W_cols = (const int*)  d_in[3];
    const float* B      = (const float*)d_in[4];
    float*       y      = (float*)d_out;
    const int    nnz    = in_sizes[1];

    SparseLinear_init_y<<<(OUT_DIMS + 255) / 256, 256, 0, stream>>>(B, y);

    long long nWin = ((long long)nnz + 127) >> 7;
    long long b    = (nWin + 7) / 8;          // 8 waves per 256-thread block
    int blocks = (b < 1) ? 1 : (b > 2048 ? 2048 : (int)b);
    SparseLinear_spmv_coo<<<blocks, 256, 0, stream>>>(W_vals, W_rows, W_cols, x, y, nnz);
}